// MyLSTM_80616536146774
// MI455X (gfx1250) — compile-verified
//
#include <hip/hip_runtime.h>

// ---------------------------------------------------------------------------
// LSTM (H=4, I=4) over T=512 for B=16384 sequences + final FC -> [B,1].
// Gate GEMM via V_WMMA_F32_16X16X4_F32 (K=4 == I == H).
// One wave (32 lanes) handles 32 batch columns: two 16-col WMMA tiles.
// Tile-1 A matrix has gate rows rotated by 8 ([g,o,i,f]) so that after one
// cndmask + permlanex16 per accumulator VGPR, every lane uniformly holds
// (i,f) locally and (g,o) from the opposite half. Sigmoid 0.5-prescale is
// folded into the i/f/o weight rows. X is staged to LDS with
// GLOBAL_LOAD_ASYNC_TO_LDS_B128 (ASYNCcnt) when available.
// ---------------------------------------------------------------------------

typedef float v2f __attribute__((ext_vector_type(2)));
typedef float v4f __attribute__((ext_vector_type(4)));
typedef float v8f __attribute__((ext_vector_type(8)));
typedef int   v4i __attribute__((ext_vector_type(4)));

#define B_TOTAL 16384
#define T_STEPS 512
#define TCHUNK  64          // timesteps staged in LDS per chunk
#define COLS    32          // batch columns per wave/block

#if __has_builtin(__builtin_amdgcn_global_load_async_to_lds_b128)
#define HAVE_ASYNC_LDS 1
#endif

__device__ __forceinline__ float tanh_fast(float x) {
#if __has_builtin(__builtin_amdgcn_tanhf)
  return __builtin_amdgcn_tanhf(x);          // v_tanh_f32 (gfx1250 trans op)
#else
  float r;
  asm volatile("v_tanh_f32 %0, %1" : "=v"(r) : "v"(x));
  return r;
#endif
}

// Swap lower/upper 16 lanes of the wave (VALU, no DScnt traffic).
__device__ __forceinline__ float swap16(float x) {
#if __has_builtin(__builtin_amdgcn_permlanex16)
  int xi = __float_as_int(x);
  int r = __builtin_amdgcn_permlanex16(xi, xi, 0x76543210, 0xfedcba98,
                                       false, false);
  return __int_as_float(r);
#else
  int r = __builtin_amdgcn_ds_swizzle(__float_as_int(x), 0x401F); // SWAPX16
  return __int_as_float(r);
#endif
}

// Half-wave select in ONE instruction: lanes 0-15 -> a, lanes 16-31 -> b.
// v_cndmask_b32 dst, src0, src1, ssrc2 : dst = ssrc2[lane] ? src1 : src0
__device__ __forceinline__ float sel_lo(float a, float b) {
  float r;
  asm("v_cndmask_b32 %0, %1, %2, %3"
      : "=v"(r)
      : "v"(b), "v"(a), "s"(0x0000FFFFu));
  return r;
}

__device__ __forceinline__ void wait_async_lds() {
#ifdef HAVE_ASYNC_LDS
#if __has_builtin(__builtin_amdgcn_s_wait_asynccnt)
  __builtin_amdgcn_s_wait_asynccnt(0);
#else
  asm volatile("s_wait_asynccnt 0x0" ::: "memory");
#endif
#endif
}

__global__ __launch_bounds__(32)
void lstm_wmma_kernel(const float* __restrict__ X,
                      const float* __restrict__ W_ih,
                      const float* __restrict__ W_hh,
                      const float* __restrict__ b_ih,
                      const float* __restrict__ b_hh,
                      const float* __restrict__ W_fc,
                      const float* __restrict__ b_fc,
                      float* __restrict__ out) {
  __shared__ float lds[COLS * TCHUNK * 4];   // [col][tt][4] f32

  const int lane  = threadIdx.x;             // wave32: 0..31
  const int b0    = blockIdx.x * COLS;
  const int m     = lane & 15;               // A-matrix row / tile column id
  const int khalf = lane >> 4;               // 0: lanes 0-15, 1: lanes 16-31

  // ---- A operands (weights), constant across the whole sequence ----------
  // 16x4 f32 A layout: lanes 0-15 hold K=0,1 ; lanes 16-31 hold K=2,3.
  // A0: natural gate-row order [i,f,g,o]. A1: rows rotated by 8 -> [g,o,i,f].
  // i/f/o rows (and biases) pre-scaled by 0.5 for sigmoid-via-tanh.
  const int w0 = m;                 // W row held by A0 row m
  const int w1 = (m + 8) & 15;      // W row held by A1 row m
  const float s0 = (w0 >= 8 && w0 < 12) ? 1.0f : 0.5f;   // g rows unscaled
  const float s1 = (w1 >= 8 && w1 < 12) ? 1.0f : 0.5f;

  v2f a_ih0, a_hh0, a_ih1, a_hh1;
  a_ih0.x = W_ih[w0 * 4 + khalf * 2 + 0] * s0;
  a_ih0.y = W_ih[w0 * 4 + khalf * 2 + 1] * s0;
  a_hh0.x = W_hh[w0 * 4 + khalf * 2 + 0] * s0;
  a_hh0.y = W_hh[w0 * 4 + khalf * 2 + 1] * s0;
  a_ih1.x = W_ih[w1 * 4 + khalf * 2 + 0] * s1;
  a_ih1.y = W_ih[w1 * 4 + khalf * 2 + 1] * s1;
  a_hh1.x = W_hh[w1 * 4 + khalf * 2 + 0] * s1;
  a_hh1.y = W_hh[w1 * 4 + khalf * 2 + 1] * s1;

  // ---- biases as initial accumulators (row-broadcast over columns) -------
  // D layout: lane L, VGPR v -> matrix row M = v + 8*(L>=16)
  v8f cbias0, cbias1;
#pragma unroll
  for (int v = 0; v < 8; ++v) {
    int g0 = v + khalf * 8;          // gate row for acc0
    int g1 = v + (khalf ^ 1) * 8;    // gate row for acc1 (rotated A)
    float t0 = (g0 >= 8 && g0 < 12) ? 1.0f : 0.5f;
    float t1 = (g1 >= 8 && g1 < 12) ? 1.0f : 0.5f;
    cbias0[v] = (b_ih[g0] + b_hh[g0]) * t0;
    cbias1[v] = (b_ih[g1] + b_hh[g1]) * t1;
  }

  // ---- per-lane LDS read offsets (float indices) --------------------------
  const int o0 = (m * TCHUNK) * 4 + khalf;          // tile0: col = m
  const int o1 = ((m + 16) * TCHUNK) * 4 + khalf;   // tile1: col = m + 16

  float hv[4] = {0.f, 0.f, 0.f, 0.f};
  float cv[4] = {0.f, 0.f, 0.f, 0.f};

  const v4f* Xv   = (const v4f*)X;          // one v4f per (b, t)
  v4f*       ldsv = (v4f*)lds;

  for (int t0 = 0; t0 < T_STEPS; t0 += TCHUNK) {
    // ---- stage chunk into LDS, coalesced --------------------------------
    // Preferred: async memory->LDS DMA (no VGPR round-trip, ASYNCcnt).
#pragma unroll 4
    for (int r = 0; r < (COLS * TCHUNK) / 32; ++r) {
      int idx = r * 32 + lane;               // [col][tt] flattened
      int col = idx >> 6;                    // TCHUNK == 64
      int tt  = idx & (TCHUNK - 1);
      size_t gidx = (size_t)(b0 + col) * T_STEPS + t0 + tt;
#ifdef HAVE_ASYNC_LDS
      __builtin_amdgcn_global_load_async_to_lds_b128(
          (v4i*)&Xv[gidx], (v4i*)&ldsv[idx], 0, 0);
#else
      ldsv[idx] = __builtin_nontemporal_load(&Xv[gidx]);
#endif
    }
    wait_async_lds();
    __syncthreads();

#pragma unroll 2
    for (int tt = 0; tt < TCHUNK; ++tt) {
      const int base = tt * 4;
      // B operands for the X part (pairs merge into ds_load_2addr_b32)
      float bx0_0 = lds[o0 + base];
      float bx0_1 = lds[o0 + base + 2];
      float bx1_0 = lds[o1 + base];
      float bx1_1 = lds[o1 + base + 2];

      // B operands for the h part: lane j owns h[col j, 0..3]
      float s01 = swap16(sel_lo(hv[1], hv[0]));
      float s23 = swap16(sel_lo(hv[3], hv[2]));
      float bh0_0 = sel_lo(hv[0], s01);
      float bh1_0 = sel_lo(s01, hv[1]);
      float bh0_1 = sel_lo(hv[2], s23);
      float bh1_1 = sel_lo(s23, hv[3]);

      v2f bx0 = {bx0_0, bx0_1}, bx1 = {bx1_0, bx1_1};
      v2f bh0 = {bh0_0, bh0_1}, bh1 = {bh1_0, bh1_1};

      // gates = W*x + W*h + bias  (two chained K=4 f32 WMMAs per tile)
      v8f acc0 = cbias0, acc1 = cbias1;
      acc0 = __builtin_amdgcn_wmma_f32_16x16x4_f32(false, a_ih0, false, bx0,
                                                   (short)0, acc0, false, false);
      acc0 = __builtin_amdgcn_wmma_f32_16x16x4_f32(false, a_hh0, false, bh0,
                                                   (short)0, acc0, false, false);
      acc1 = __builtin_amdgcn_wmma_f32_16x16x4_f32(false, a_ih1, false, bx1,
                                                   (short)0, acc1, false, false);
      acc1 = __builtin_amdgcn_wmma_f32_16x16x4_f32(false, a_hh1, false, bh1,
                                                   (short)0, acc1, false, false);

      // Uniform redistribution (thanks to the rotated A1):
      //   L[u] = lanes' own (i,f) rows ; S[u] = (g,o) rows from other half.
      float L[8], S[8];
#pragma unroll
      for (int u = 0; u < 8; ++u) {
        L[u] = sel_lo(acc0[u], acc1[u]);
        S[u] = swap16(sel_lo(acc1[u], acc0[u]));
      }

#pragma unroll
      for (int u = 0; u < 4; ++u) {
        float i_g = __builtin_fmaf(tanh_fast(L[u]),     0.5f, 0.5f);
        float f_g = __builtin_fmaf(tanh_fast(L[u + 4]), 0.5f, 0.5f);
        float g_g = tanh_fast(S[u]);
        float o_g = __builtin_fmaf(tanh_fast(S[u + 4]), 0.5f, 0.5f);
        cv[u] = __builtin_fmaf(f_g, cv[u], i_g * g_g);
        hv[u] = o_g * tanh_fast(cv[u]);
      }
    }
    __syncthreads();
  }

  // ---- final FC: out[b] = h_last . W_fc + b_fc (lane owns column = lane) --
  float r = b_fc[0];
  r = __builtin_fmaf(hv[0], W_fc[0], r);
  r = __builtin_fmaf(hv[1], W_fc[1], r);
  r = __builtin_fmaf(hv[2], W_fc[2], r);
  r = __builtin_fmaf(hv[3], W_fc[3], r);
  out[b0 + lane] = r;
}

extern "C" void kernel_launch(void* const* d_in, const int* in_sizes, int n_in,
                              void* d_out, int out_size, void* d_ws, size_t ws_size,
                              hipStream_t stream) {
  (void)in_sizes; (void)n_in; (void)d_ws; (void)ws_size; (void)out_size;
  const float* X    = (const float*)d_in[0];
  const float* W_ih = (const float*)d_in[1];
  const float* W_hh = (const float*)d_in[2];
  const float* b_ih = (const float*)d_in[3];
  const float* b_hh = (const float*)d_in[4];
  const float* W_fc = (const float*)d_in[5];
  const float* b_fc = (const float*)d_in[6];
  float* out = (float*)d_out;

  dim3 grid(B_TOTAL / COLS);   // 512 single-wave workgroups
  dim3 block(32);
  lstm_wmma_kernel<<<grid, block, 0, stream>>>(X, W_ih, W_hh, b_ih, b_hh,
                                               W_fc, b_fc, out);
}